// SimpleGCN_2843268350877
// MI455X (gfx1250) — compile-verified
//
#include <hip/hip_runtime.h>

#define NN 100000
#define EE 1600000
#define FIN 128

typedef float v2f __attribute__((ext_vector_type(2)));
typedef float v8f __attribute__((ext_vector_type(8)));

__device__ __forceinline__ v8f splat8(float v) {
  v8f r = {v, v, v, v, v, v, v, v};
  return r;
}

// float atomic max via monotone int mapping (native global_atomic_max_i32 /
// global_atomic_min_u32, no CAS loop). Init value must be -inf (0xFF800000).
__device__ __forceinline__ void atomicMaxF(float* addr, float val) {
  if (val >= 0.0f) atomicMax((int*)addr, __float_as_int(val));
  else             atomicMin((unsigned int*)addr, __float_as_uint(val));
}

// ---------------------------------------------------------------------------
// Layer-1 fused projections: qkvs[row][0:16]=q, [16:32]=k, [32:48]=v,
// [48:64]=skip.  One pass over x, 4 accumulators, V_WMMA_F32_16X16X4_F32.
// f32 A layout (ISA 7.12.2): lanes 0-15 -> K = {k0, k0+1}; lanes 16-31 ->
// K = {k0+2, k0+3}; row M = lane&15. B mirrors (rows over VGPR/half-wave,
// N over lanes). C/D: VGPR r = row r (lanes<16) / row r+8 (lanes>=16).
// ---------------------------------------------------------------------------
__global__ __launch_bounds__(256) void gemm1_wmma(
    const float* __restrict__ x,
    const float* __restrict__ Wq, const float* __restrict__ bq,
    const float* __restrict__ Wk, const float* __restrict__ bk,
    const float* __restrict__ Wv, const float* __restrict__ bv,
    const float* __restrict__ Ws, const float* __restrict__ bs,
    float* __restrict__ out) {
  const int lane = threadIdx.x & 31;
  const int wave = threadIdx.x >> 5;
  const int hlf  = lane >> 4;
  const int m    = lane & 15;
  const int rowBase = blockIdx.x * 128 + wave * 16;
  if (rowBase >= NN) return;  // wave-uniform exit: EXEC stays all-ones for WMMA

  const float* xrow = x + (size_t)min(rowBase + m, NN - 1) * FIN;

  v8f cq = splat8(bq[m]);
  v8f ck = splat8(bk[m]);
  v8f cv = splat8(bv[m]);
  v8f cs = splat8(bs[m]);

  for (int k0 = 0; k0 < FIN; k0 += 4) {
    const int ka = k0 + 2 * hlf;
    v2f a;
    a.x = xrow[ka];
    a.y = xrow[ka + 1];
    const int kr0 = ka * 16 + m;
    const int kr1 = kr0 + 16;
    v2f b;
    b.x = Wq[kr0]; b.y = Wq[kr1];
    cq = __builtin_amdgcn_wmma_f32_16x16x4_f32(false, a, false, b, (short)0, cq, false, false);
    b.x = Wk[kr0]; b.y = Wk[kr1];
    ck = __builtin_amdgcn_wmma_f32_16x16x4_f32(false, a, false, b, (short)0, ck, false, false);
    b.x = Wv[kr0]; b.y = Wv[kr1];
    cv = __builtin_amdgcn_wmma_f32_16x16x4_f32(false, a, false, b, (short)0, cv, false, false);
    b.x = Ws[kr0]; b.y = Ws[kr1];
    cs = __builtin_amdgcn_wmma_f32_16x16x4_f32(false, a, false, b, (short)0, cs, false, false);
  }

#pragma unroll
  for (int r = 0; r < 8; ++r) {
    const int row = rowBase + r + 8 * hlf;
    if (row < NN) {
      float* o = out + (size_t)row * 64 + m;
      o[0]  = cq[r];
      o[16] = ck[r];
      o[32] = cv[r];
      o[48] = cs[r];
    }
  }
}

// ---------------------------------------------------------------------------
// Layer-2 fused projections: h[N,16] @ [16,10]x4 -> qkvs2[N,48]:
// q@0, k@10, v@20, skip@30 (cols 40..47 unused). K=16 -> 4 WMMA steps.
// Lanes with m>=10 contribute zero B columns (branchless mask, EXEC intact).
// ---------------------------------------------------------------------------
__global__ __launch_bounds__(256) void gemm2_wmma(
    const float* __restrict__ h,
    const float* __restrict__ Wq, const float* __restrict__ bq,
    const float* __restrict__ Wk, const float* __restrict__ bk,
    const float* __restrict__ Wv, const float* __restrict__ bv,
    const float* __restrict__ Ws, const float* __restrict__ bs,
    float* __restrict__ out) {
  const int lane = threadIdx.x & 31;
  const int wave = threadIdx.x >> 5;
  const int hlf  = lane >> 4;
  const int m    = lane & 15;
  const int rowBase = blockIdx.x * 128 + wave * 16;
  if (rowBase >= NN) return;

  const float* hrow = h + (size_t)min(rowBase + m, NN - 1) * 16;
  const int   j    = (m < 10) ? m : 0;
  const float msk  = (m < 10) ? 1.0f : 0.0f;

  v8f cq = splat8(bq[j] * msk);
  v8f ck = splat8(bk[j] * msk);
  v8f cv = splat8(bv[j] * msk);
  v8f cs = splat8(bs[j] * msk);

#pragma unroll
  for (int k0 = 0; k0 < 16; k0 += 4) {
    const int ka = k0 + 2 * hlf;
    v2f a;
    a.x = hrow[ka];
    a.y = hrow[ka + 1];
    const int kr0 = ka * 10 + j;
    const int kr1 = kr0 + 10;
    v2f b;
    b.x = Wq[kr0] * msk; b.y = Wq[kr1] * msk;
    cq = __builtin_amdgcn_wmma_f32_16x16x4_f32(false, a, false, b, (short)0, cq, false, false);
    b.x = Wk[kr0] * msk; b.y = Wk[kr1] * msk;
    ck = __builtin_amdgcn_wmma_f32_16x16x4_f32(false, a, false, b, (short)0, ck, false, false);
    b.x = Wv[kr0] * msk; b.y = Wv[kr1] * msk;
    cv = __builtin_amdgcn_wmma_f32_16x16x4_f32(false, a, false, b, (short)0, cv, false, false);
    b.x = Ws[kr0] * msk; b.y = Ws[kr1] * msk;
    cs = __builtin_amdgcn_wmma_f32_16x16x4_f32(false, a, false, b, (short)0, cs, false, false);
  }

#pragma unroll
  for (int r = 0; r < 8; ++r) {
    const int row = rowBase + r + 8 * hlf;
    if (row < NN && m < 10) {
      float* o = out + (size_t)row * 48 + m;
      o[0]  = cq[r];
      o[10] = ck[r];
      o[20] = cv[r];
      o[30] = cs[r];
    }
  }
}

// ---------------------------------------------------------------------------
// Segment-softmax scaffolding
// ---------------------------------------------------------------------------
__global__ void seg_init(float* __restrict__ mx, float* __restrict__ dn,
                         float* __restrict__ acc) {
  const int i = blockIdx.x * 256 + threadIdx.x;
  if (i >= NN) return;
  mx[i] = -__builtin_inff();
  dn[i] = 0.0f;
  float4 z = make_float4(0.f, 0.f, 0.f, 0.f);
  float4* a = (float4*)(acc + (size_t)i * 16);
  a[0] = z; a[1] = z; a[2] = z; a[3] = z;
}

__global__ void edge_score1(const int* __restrict__ ei,
                            const float* __restrict__ qkvs,
                            float* __restrict__ score, float* __restrict__ mx) {
  const int e = blockIdx.x * 256 + threadIdx.x;
  if (e >= EE) return;
  const int s = ei[e], d = ei[EE + e];
  const float4* q = (const float4*)(qkvs + (size_t)d * 64);
  const float4* k = (const float4*)(qkvs + (size_t)s * 64 + 16);
  float acc = 0.0f;
#pragma unroll
  for (int i = 0; i < 4; ++i) {
    float4 qa = q[i], kb = k[i];
    acc += qa.x * kb.x + qa.y * kb.y + qa.z * kb.z + qa.w * kb.w;
  }
  acc *= 0.25f;  // 1/sqrt(16)
  score[e] = acc;
  atomicMaxF(&mx[d], acc);
}

__global__ void edge_accum1(const int* __restrict__ ei,
                            const float* __restrict__ qkvs,
                            const float* __restrict__ score,
                            const float* __restrict__ mx,
                            float* __restrict__ dn, float* __restrict__ acc) {
  const int e = blockIdx.x * 256 + threadIdx.x;
  if (e >= EE) return;
  const int s = ei[e], d = ei[EE + e];
  const float w = __expf(score[e] - mx[d]);
  atomicAdd(&dn[d], w);
  const float* v = qkvs + (size_t)s * 64 + 32;
  float* a = acc + (size_t)d * 16;
#pragma unroll
  for (int c = 0; c < 16; ++c) atomicAdd(a + c, w * v[c]);
}

__global__ void finalize1(const float* __restrict__ qkvs,
                          const float* __restrict__ acc,
                          const float* __restrict__ dn, float* __restrict__ h) {
  const int i = blockIdx.x * 256 + threadIdx.x;
  if (i >= NN) return;
  const float d = dn[i];
  const float inv = d > 0.0f ? 1.0f / d : 0.0f;
#pragma unroll
  for (int c = 0; c < 16; ++c) {
    float val = acc[(size_t)i * 16 + c] * inv + qkvs[(size_t)i * 64 + 48 + c];
    h[(size_t)i * 16 + c] = fmaxf(val, 0.0f);  // relu
  }
}

__global__ void edge_score2(const int* __restrict__ ei,
                            const float* __restrict__ qkvs,
                            float* __restrict__ score, float* __restrict__ mx) {
  const int e = blockIdx.x * 256 + threadIdx.x;
  if (e >= EE) return;
  const int s = ei[e], d = ei[EE + e];
  const float* q = qkvs + (size_t)d * 48;
  const float* k = qkvs + (size_t)s * 48 + 10;
  float acc = 0.0f;
#pragma unroll
  for (int c = 0; c < 10; ++c) acc += q[c] * k[c];
  acc *= 0.31622776601683794f;  // 1/sqrt(10)
  score[e] = acc;
  atomicMaxF(&mx[d], acc);
}

__global__ void edge_accum2(const int* __restrict__ ei,
                            const float* __restrict__ qkvs,
                            const float* __restrict__ score,
                            const float* __restrict__ mx,
                            float* __restrict__ dn, float* __restrict__ acc) {
  const int e = blockIdx.x * 256 + threadIdx.x;
  if (e >= EE) return;
  const int s = ei[e], d = ei[EE + e];
  const float w = __expf(score[e] - mx[d]);
  atomicAdd(&dn[d], w);
  const float* v = qkvs + (size_t)s * 48 + 20;
  float* a = acc + (size_t)d * 16;
#pragma unroll
  for (int c = 0; c < 10; ++c) atomicAdd(a + c, w * v[c]);
}

__global__ void finalize2(const float* __restrict__ qkvs,
                          const float* __restrict__ acc,
                          const float* __restrict__ dn, float* __restrict__ out) {
  const int i = blockIdx.x * 256 + threadIdx.x;
  if (i >= NN) return;
  const float d = dn[i];
  const float inv = d > 0.0f ? 1.0f / d : 0.0f;
  float vals[10];
  float m = -__builtin_inff();
#pragma unroll
  for (int c = 0; c < 10; ++c) {
    float v = acc[(size_t)i * 16 + c] * inv + qkvs[(size_t)i * 48 + 30 + c];
    vals[c] = v;
    m = fmaxf(m, v);
  }
  float ss = 0.0f;
#pragma unroll
  for (int c = 0; c < 10; ++c) ss += __expf(vals[c] - m);
  const float lse = m + __logf(ss);
#pragma unroll
  for (int c = 0; c < 10; ++c) out[(size_t)i * 10 + c] = vals[c] - lse;
}

// ---------------------------------------------------------------------------
extern "C" void kernel_launch(void* const* d_in, const int* in_sizes, int n_in,
                              void* d_out, int out_size, void* d_ws, size_t ws_size,
                              hipStream_t stream) {
  (void)in_sizes; (void)n_in; (void)out_size; (void)ws_size;
  const float* x  = (const float*)d_in[0];
  const int*   ei = (const int*)d_in[1];
  const float *Wq1 = (const float*)d_in[2],  *bq1 = (const float*)d_in[3],
              *Wk1 = (const float*)d_in[4],  *bk1 = (const float*)d_in[5],
              *Wv1 = (const float*)d_in[6],  *bv1 = (const float*)d_in[7],
              *Ws1 = (const float*)d_in[8],  *bs1 = (const float*)d_in[9],
              *Wq2 = (const float*)d_in[10], *bq2 = (const float*)d_in[11],
              *Wk2 = (const float*)d_in[12], *bk2 = (const float*)d_in[13],
              *Wv2 = (const float*)d_in[14], *bv2 = (const float*)d_in[15],
              *Ws2 = (const float*)d_in[16], *bs2 = (const float*)d_in[17];
  float* out = (float*)d_out;

  // workspace layout (floats), ~45.6 MB total; qkvs region reused by layer 2
  float* ws    = (float*)d_ws;
  float* qkvs  = ws;                       // N*64 (L1) / N*48 (L2)
  float* score = qkvs + (size_t)NN * 64;   // E
  float* mx    = score + (size_t)EE;       // N
  float* dn    = mx + NN;                  // N
  float* acc   = dn + NN;                  // N*16
  float* h     = acc + (size_t)NN * 16;    // N*16

  const dim3 b(256);
  const int gRows  = (NN + 127) / 128;
  const int gNodes = (NN + 255) / 256;
  const int gEdges = (EE + 255) / 256;

  // Layer 1
  gemm1_wmma<<<gRows, b, 0, stream>>>(x, Wq1, bq1, Wk1, bk1, Wv1, bv1, Ws1, bs1, qkvs);
  seg_init<<<gNodes, b, 0, stream>>>(mx, dn, acc);
  edge_score1<<<gEdges, b, 0, stream>>>(ei, qkvs, score, mx);
  edge_accum1<<<gEdges, b, 0, stream>>>(ei, qkvs, score, mx, dn, acc);
  finalize1<<<gNodes, b, 0, stream>>>(qkvs, acc, dn, h);

  // Layer 2
  gemm2_wmma<<<gRows, b, 0, stream>>>(h, Wq2, bq2, Wk2, bk2, Wv2, bv2, Ws2, bs2, qkvs);
  seg_init<<<gNodes, b, 0, stream>>>(mx, dn, acc);
  edge_score2<<<gEdges, b, 0, stream>>>(ei, qkvs, score, mx);
  edge_accum2<<<gEdges, b, 0, stream>>>(ei, qkvs, score, mx, dn, acc);
  finalize2<<<gNodes, b, 0, stream>>>(qkvs, acc, dn, out);
}